// GraphProjection_81123342286853
// MI455X (gfx1250) — compile-verified
//
#include <hip/hip_runtime.h>
#include <math.h>

typedef __attribute__((ext_vector_type(2))) float v2f;
typedef __attribute__((ext_vector_type(4))) float v4f;
typedef __attribute__((ext_vector_type(8))) float v8f;

// Async-LDS builtin pointee type per hipcc diagnostic: int __attribute__((vector_size(16)))
typedef int v4i __attribute__((vector_size(16)));
typedef __attribute__((address_space(1))) v4i v4i_gm;
typedef __attribute__((address_space(3))) v4i v4i_lds;

#define DEGF 0.017453292519943295f
#define BLOCK 256
#define WAVES_PER_BLOCK 8
#define F4_ELEMS 25088   // 7*7*512 floats, view-0 slice of f4 (100,352 B -> LDS)

// ---------------------------------------------------------------------------
// Setup: fold camera math into one affine map per view.
// pts_v = inputs @ Bv + tv   where Bv = inv(c0)^T @ c_v^T,  tv = (o0-o_v) @ c_v^T
// ws layout per view v (12 floats): Bv[j][d] at ws[v*12 + j*3 + d], tv[d] at ws[v*12+9+d]
// ---------------------------------------------------------------------------
__global__ void cam_setup_kernel(const float* __restrict__ cams,
                                 float* __restrict__ ws) {
  if (threadIdx.x != 0 || blockIdx.x != 0) return;
  float C[3][3][3], O[3][3];
  for (int v = 0; v < 3; ++v) {
    const float* p = cams + 5 * v;
    float th = p[0] * DEGF, ph = p[1] * DEGF, dist = p[3];
    float camy = dist * sinf(ph);
    float lens = dist * cosf(ph);
    float ct = cosf(th), st = sinf(th);
    float Zv[3] = {lens * ct, camy, lens * st};
    float Yv[3] = {-camy * ct, lens, -camy * st};
    float Xv[3] = {Yv[1] * Zv[2] - Yv[2] * Zv[1],
                   Yv[2] * Zv[0] - Yv[0] * Zv[2],
                   Yv[0] * Zv[1] - Yv[1] * Zv[0]};
    float rx = rsqrtf(Xv[0] * Xv[0] + Xv[1] * Xv[1] + Xv[2] * Xv[2]);
    float ry = rsqrtf(Yv[0] * Yv[0] + Yv[1] * Yv[1] + Yv[2] * Yv[2]);
    float rz = rsqrtf(Zv[0] * Zv[0] + Zv[1] * Zv[1] + Zv[2] * Zv[2]);
    for (int k = 0; k < 3; ++k) {
      C[v][0][k] = Xv[k] * rx;
      C[v][1][k] = Yv[k] * ry;
      C[v][2][k] = Zv[k] * rz;
      O[v][k] = Zv[k];           // reference returns unnormalized Z as origin
    }
  }
  // inv(C[0]) via adjugate
  float (*c)[3] = C[0];
  float det = c[0][0] * (c[1][1] * c[2][2] - c[1][2] * c[2][1])
            - c[0][1] * (c[1][0] * c[2][2] - c[1][2] * c[2][0])
            + c[0][2] * (c[1][0] * c[2][1] - c[1][1] * c[2][0]);
  float id = 1.0f / det;
  float inv[3][3];
  inv[0][0] = (c[1][1] * c[2][2] - c[1][2] * c[2][1]) * id;
  inv[0][1] = (c[0][2] * c[2][1] - c[0][1] * c[2][2]) * id;
  inv[0][2] = (c[0][1] * c[1][2] - c[0][2] * c[1][1]) * id;
  inv[1][0] = (c[1][2] * c[2][0] - c[1][0] * c[2][2]) * id;
  inv[1][1] = (c[0][0] * c[2][2] - c[0][2] * c[2][0]) * id;
  inv[1][2] = (c[0][2] * c[1][0] - c[0][0] * c[1][2]) * id;
  inv[2][0] = (c[1][0] * c[2][1] - c[1][1] * c[2][0]) * id;
  inv[2][1] = (c[0][1] * c[2][0] - c[0][0] * c[2][1]) * id;
  inv[2][2] = (c[0][0] * c[1][1] - c[0][1] * c[1][0]) * id;
  // A0[j][k] = inv[k][j]   (inv(c0)^T)
  for (int v = 0; v < 3; ++v) {
    for (int j = 0; j < 3; ++j)
      for (int d = 0; d < 3; ++d) {
        float s = 0.f;
        for (int k = 0; k < 3; ++k) s += inv[k][j] * C[v][d][k];
        ws[v * 12 + j * 3 + d] = s;
      }
    for (int d = 0; d < 3; ++d) {
      float s = 0.f;
      for (int k = 0; k < 3; ++k) s += (O[0][k] - O[v][k]) * C[v][d][k];
      ws[v * 12 + 9 + d] = s;
    }
  }
}

// ---------------------------------------------------------------------------
// Main kernel: one wave per 16-point group.
//   WMMA 16x16x4 f32: A = 16 pts x (x,y,z,1); B cols 0..8 = 3 views x 3 dims.
//   Then gather 960 channels/view (f4 slice staged in LDS) + max/mean/std.
// ---------------------------------------------------------------------------
__global__ __launch_bounds__(BLOCK)
void graphproj_kernel(const float* __restrict__ pts_in,
                      const float* __restrict__ f1,
                      const float* __restrict__ f2,
                      const float* __restrict__ f3,
                      const float* __restrict__ f4,
                      const float* __restrict__ ws,
                      float* __restrict__ out, int N) {
  __shared__ float s_f4[F4_ELEMS];
  __shared__ float s_pts[WAVES_PER_BLOCK][16][9];

  const int tid = threadIdx.x;

  // ---- stage f4 view-0 slice into LDS (async tensor/DMA path) ----
#if __has_builtin(__builtin_amdgcn_global_load_async_to_lds_b128)
  for (int i = tid; i < F4_ELEMS / 4; i += BLOCK) {
    __builtin_amdgcn_global_load_async_to_lds_b128(
        (v4i_gm*)(unsigned long long)(const void*)(f4 + 4 * i),
        (v4i_lds*)(unsigned int)(unsigned long long)(const void*)(s_f4 + 4 * i),
        0, 0);
  }
#if __has_builtin(__builtin_amdgcn_s_wait_asynccnt)
  __builtin_amdgcn_s_wait_asynccnt(0);
#else
  asm volatile("s_wait_asynccnt 0" ::: "memory");
#endif
#else
  for (int i = tid; i < F4_ELEMS / 4; i += BLOCK)
    *(v4f*)(s_f4 + 4 * i) = *(const v4f*)(f4 + 4 * i);
#endif

  const int wave = tid >> 5;
  const int lane = tid & 31;
  const int half = lane >> 4;   // 0: K=0,1 / M=0..7 ; 1: K=2,3 / M=8..15
  const int n    = lane & 15;
  const int grp  = blockIdx.x * WAVES_PER_BLOCK + wave;
  const int base = grp * 16;

  // ---- WMMA point transform ----
#if __has_builtin(__builtin_amdgcn_wmma_f32_16x16x4_f32)
  {
    int prow = base + n;                      // A row M = n
    v2f a; a.x = 0.f; a.y = 0.f;
    if (prow < N) {
      if (half == 0) { a.x = pts_in[3 * prow];     a.y = pts_in[3 * prow + 1]; }
      else           { a.x = pts_in[3 * prow + 2]; a.y = 1.0f; }
    }
    v2f b; b.x = 0.f; b.y = 0.f;              // B col N = n (cols 9..15 zero)
    if (n < 9) {
      int v = n / 3, d = n - 3 * v;
      const float* wv = ws + v * 12;
      if (half == 0) { b.x = wv[d];     b.y = wv[3 + d]; }   // K=0,1
      else           { b.x = wv[6 + d]; b.y = wv[9 + d]; }   // K=2,3 (t row)
    }
    v8f acc = {};
    acc = __builtin_amdgcn_wmma_f32_16x16x4_f32(
        false, a, false, b, (short)0, acc, false, false);
    if (n < 9) {
#pragma unroll
      for (int i = 0; i < 8; ++i)
        s_pts[wave][i + 8 * half][n] = acc[i];   // D[M=i+8*half][N=n]
    }
  }
#else
  if (half == 0) {   // scalar fallback, lanes 0..15
    int p = base + n;
    float x = 0.f, y = 0.f, z = 0.f;
    if (p < N) { x = pts_in[3 * p]; y = pts_in[3 * p + 1]; z = pts_in[3 * p + 2]; }
    for (int v = 0; v < 3; ++v)
      for (int d = 0; d < 3; ++d)
        s_pts[wave][n][3 * v + d] = x * ws[v * 12 + d] + y * ws[v * 12 + 3 + d] +
                                    z * ws[v * 12 + 6 + d] + ws[v * 12 + 9 + d];
  }
#endif

  __syncthreads();
  if (base >= N) return;

  // ---- per-point gather + reduce over views ----
  for (int pl = 0; pl < 16; ++pl) {
    int p = base + pl;
    if (p >= N) break;
    int b1[3], b2[3], b3[3], b4[3];
#pragma unroll
    for (int v = 0; v < 3; ++v) {
      float X = s_pts[wave][pl][3 * v + 0];
      float Y = s_pts[wave][pl][3 * v + 1];
      float Z = s_pts[wave][pl][3 * v + 2];
      float rz = 1.0f / Z;
      float h  = fminf(fmaxf( 248.0f * Y * rz + 112.0f, 0.0f), 223.0f);
      float wc = fminf(fmaxf(-248.0f * X * rz + 112.0f, 0.0f), 223.0f);
      b1[v] = ((int)(h * 0.25f)    * 56 + (int)(wc * 0.25f))    * 64;
      b2[v] = ((int)(h * 0.125f)   * 28 + (int)(wc * 0.125f))   * 128;
      b3[v] = ((int)(h * 0.0625f)  * 14 + (int)(wc * 0.0625f))  * 256;
      b4[v] = ((int)(h * 0.03125f) * 7  + (int)(wc * 0.03125f)) * 512;
      __builtin_prefetch(f3 + b3[v], 0, 1);   // global_prefetch_b8
    }
    size_t ob = (size_t)p * 2883;
    if (lane < 3) out[ob + lane] = pts_in[3 * p + lane];

    for (int c = lane * 4; c < 960; c += 128) {
      v4f g[3];
#pragma unroll
      for (int v = 0; v < 3; ++v) {
        if      (c < 64)  g[v] = *(const v4f*)(f1   + b1[v] + c);
        else if (c < 192) g[v] = *(const v4f*)(f2   + b2[v] + (c - 64));
        else if (c < 448) g[v] = *(const v4f*)(f3   + b3[v] + (c - 192));
        else              g[v] = *(const v4f*)(s_f4 + b4[v] + (c - 448));
      }
      v4f mx;
      mx.x = fmaxf(fmaxf(g[0].x, g[1].x), g[2].x);
      mx.y = fmaxf(fmaxf(g[0].y, g[1].y), g[2].y);
      mx.z = fmaxf(fmaxf(g[0].z, g[1].z), g[2].z);
      mx.w = fmaxf(fmaxf(g[0].w, g[1].w), g[2].w);
      v4f mean = (g[0] + g[1] + g[2]) * (1.0f / 3.0f);
      v4f d0 = g[0] - mean, d1 = g[1] - mean, d2 = g[2] - mean;
      v4f var = (d0 * d0 + d1 * d1 + d2 * d2) * (1.0f / 3.0f);
      v4f sd;
      sd.x = sqrtf(var.x); sd.y = sqrtf(var.y);
      sd.z = sqrtf(var.z); sd.w = sqrtf(var.w);
      // rows are only 4B-aligned (stride 2883): let the backend pick the store width
      __builtin_memcpy(out + ob + 3 + c,    &mx,   16);
      __builtin_memcpy(out + ob + 963 + c,  &mean, 16);
      __builtin_memcpy(out + ob + 1923 + c, &sd,   16);
    }
  }
}

// ---------------------------------------------------------------------------
extern "C" void kernel_launch(void* const* d_in, const int* in_sizes, int n_in,
                              void* d_out, int out_size, void* d_ws, size_t ws_size,
                              hipStream_t stream) {
  (void)n_in; (void)out_size; (void)ws_size;
  const float* pts  = (const float*)d_in[0];
  const float* cams = (const float*)d_in[1];
  const float* f1   = (const float*)d_in[2];
  const float* f2   = (const float*)d_in[3];
  const float* f3   = (const float*)d_in[4];
  const float* f4   = (const float*)d_in[5];
  float* out = (float*)d_out;
  float* ws  = (float*)d_ws;

  int N = in_sizes[0] / 3;
  cam_setup_kernel<<<1, 32, 0, stream>>>(cams, ws);
  int groups = (N + 15) / 16;
  int blocks = (groups + WAVES_PER_BLOCK - 1) / WAVES_PER_BLOCK;
  graphproj_kernel<<<blocks, BLOCK, 0, stream>>>(pts, f1, f2, f3, f4, ws, out, N);
}